// LossVariance_3075196584102
// MI455X (gfx1250) — compile-verified
//
#include <hip/hip_runtime.h>
#include <stdint.h>

// Problem constants (match reference setup_inputs)
#define Bn 16
#define Cn 32
#define Ln 64
#define Hn 512
#define Wn 512
#define Pn (Hn * Wn)                        // 262144 pixels per image

// Tiling
#define WGS_PER_B 128                       // workgroups per batch image
#define PIX_PER_WG (Pn / WGS_PER_B)         // 2048
#define WAVES 4                             // waves per workgroup (wave32)
#define BLOCK (WAVES * 32)                  // 128 threads
#define TILEP 64                            // pixels per tile (per wave): b64 x 32 lanes
#define TSTRIDE 66                          // padded LDS dword stride per channel row
                                            //   write: (c*66 + 2*lane)*4 -> 8B aligned, contiguous per c
                                            //   read : (lane*66 + j)*4   -> banks 2*lane, conflict-free
#define TILES_PER_WAVE (PIX_PER_WG / (WAVES * TILEP))  // 8
#define TILE_DW (Cn * TSTRIDE)              // 2112 dwords per tile buffer

#define WS_CNT   (Bn * Ln)                  // 1024 floats
#define WS_MOM   (Bn * Ln * Cn)             // 32768 floats
#define WS_TOTAL (WS_CNT + 2 * WS_MOM)      // 66560 floats

// ---------------------------------------------------------------------------
// Kernel 0: zero the workspace accumulators (ws is not re-poisoned between
// timed replays, so we must re-init every launch).
// ---------------------------------------------------------------------------
__global__ void lv_zero(float* __restrict__ ws, int n) {
    int i = blockIdx.x * blockDim.x + threadIdx.x;
    if (i < n) ws[i] = 0.0f;
}

// ---------------------------------------------------------------------------
// Kernel 1: segmented accumulation of cnt / sum / sumsq per (b, label, chan).
// Double-buffered async staging: issue tile k+1's 32 b64 async-to-LDS loads,
// s_wait_asynccnt <= 32 (async loads complete in order => tile k landed),
// then accumulate tile k transposed (lanes over channels) with conflict-free
// ds_add_f32 atomics. Labels are made wave-uniform via v_readlane.
// ---------------------------------------------------------------------------
__global__ __launch_bounds__(BLOCK) void lv_accum(
    const float* __restrict__ x, const int* __restrict__ t,
    float* __restrict__ gcnt, float* __restrict__ gsum, float* __restrict__ gss)
{
    __shared__ float s_sum[Ln * Cn];                 // 8 KB
    __shared__ float s_ss [Ln * Cn];                 // 8 KB
    __shared__ float s_cnt[Ln];                      // 256 B
    __shared__ float s_tile[WAVES][2][TILE_DW];      // 4 * 2 * 8448 B = 66 KB

    const int tid  = threadIdx.x;
    const int lane = tid & 31;
    const int wave = tid >> 5;
    const int b     = blockIdx.x / WGS_PER_B;
    const int chunk = blockIdx.x % WGS_PER_B;

    // zero LDS accumulators
    for (int i = tid; i < Ln * Cn; i += BLOCK) { s_sum[i] = 0.0f; s_ss[i] = 0.0f; }
    if (tid < Ln) s_cnt[tid] = 0.0f;
    __syncthreads();

    const unsigned long long xbase =
        (unsigned long long)(x + (size_t)b * Cn * Pn);     // SGPR-pair base for async loads
    const int* tb = t + (size_t)b * Pn;
    const unsigned tbase0 = (unsigned)(uintptr_t)(&s_tile[wave][0][0]);
    const unsigned tbase1 = (unsigned)(uintptr_t)(&s_tile[wave][1][0]);

    const int pstart = chunk * PIX_PER_WG + wave * (TILES_PER_WAVE * TILEP);
    const unsigned lane8 = (unsigned)(lane * 8);

    // -------- issue tile 0 into buffer 0 --------
    {
        unsigned p0b4 = (unsigned)(pstart * 4);
#pragma unroll
        for (int c = 0; c < Cn; ++c) {
            unsigned voff  = (unsigned)(c * Pn * 4) + p0b4 + lane8;
            unsigned laddr = tbase0 + (unsigned)(c * TSTRIDE * 4) + lane8;
            asm volatile("global_load_async_to_lds_b64 %0, %1, %2"
                         :: "v"(laddr), "v"(voff), "s"(xbase) : "memory");
        }
    }

    for (int it = 0; it < TILES_PER_WAVE; ++it) {
        const int p0 = pstart + it * TILEP;
        const unsigned curbase = (it & 1) ? tbase1 : tbase0;
        const float* curtile = &s_tile[wave][it & 1][0];

        // labels for this tile's 64 pixels (coalesced) + counts
        int lab0 = tb[p0 + lane];
        int lab1 = tb[p0 + 32 + lane];
        atomicAdd(&s_cnt[lab0], 1.0f);
        atomicAdd(&s_cnt[lab1], 1.0f);

        // -------- prefetch tile it+1 into the other buffer --------
        if (it + 1 < TILES_PER_WAVE) {
            unsigned p1b4 = (unsigned)((p0 + TILEP) * 4);
            unsigned nxtbase = (it & 1) ? tbase0 : tbase1;
#pragma unroll
            for (int c = 0; c < Cn; ++c) {
                unsigned voff  = (unsigned)(c * Pn * 4) + p1b4 + lane8;
                unsigned laddr = nxtbase + (unsigned)(c * TSTRIDE * 4) + lane8;
                asm volatile("global_load_async_to_lds_b64 %0, %1, %2"
                             :: "v"(laddr), "v"(voff), "s"(xbase) : "memory");
            }
            // in-order completion: <=32 outstanding => current tile landed
            asm volatile("s_wait_asynccnt 32" ::: "memory");
        } else {
            asm volatile("s_wait_asynccnt 0" ::: "memory");
        }

        // -------- transposed accumulate: lane == channel, loop pixels --------
#pragma unroll
        for (int h = 0; h < 2; ++h) {
            int labs = h ? lab1 : lab0;
#pragma unroll
            for (int j = 0; j < 32; ++j) {
                int l = __builtin_amdgcn_readlane(labs, j);   // wave-uniform label (SGPR)
                float v = curtile[lane * TSTRIDE + h * 32 + j];
                atomicAdd(&s_sum[l * Cn + lane], v);          // ds_add_f32, conflict-free
                atomicAdd(&s_ss [l * Cn + lane], v * v);
            }
        }
    }
    __syncthreads();

    // flush workgroup partials to global accumulators (native f32 atomics)
    const size_t mbase = (size_t)b * Ln * Cn;
    for (int i = tid; i < Ln * Cn; i += BLOCK) {
        atomicAdd(&gsum[mbase + i], s_sum[i]);
        atomicAdd(&gss [mbase + i], s_ss[i]);
    }
    if (tid < Ln) atomicAdd(&gcnt[b * Ln + tid], s_cnt[tid]);
}

// ---------------------------------------------------------------------------
// Kernel 2: finalize. One thread per (b, label) segment; reduce over labels
// per batch; mean over batches.
// ---------------------------------------------------------------------------
__global__ __launch_bounds__(Bn * Ln) void lv_final(
    const float* __restrict__ gcnt, const float* __restrict__ gsum,
    const float* __restrict__ gss, float* __restrict__ out)
{
    __shared__ float s_var [Bn * Ln];
    __shared__ float s_pres[Bn * Ln];

    const int tid = threadIdx.x;         // = b*64 + l
    const int l   = tid & (Ln - 1);

    float cnt = gcnt[tid];
    float vs = 0.0f;
    if (cnt > 1.0f) {
        float inv  = 1.0f / cnt;
        float invm = 1.0f / (cnt - 1.0f);
        const float* sp = gsum + (size_t)tid * Cn;
        const float* qp = gss  + (size_t)tid * Cn;
#pragma unroll
        for (int c = 0; c < Cn; ++c) {
            float s = sp[c];
            vs += (qp[c] - s * s * inv) * invm;
        }
    }
    s_var [tid] = (l != 0) ? vs : 0.0f;
    s_pres[tid] = (l != 0 && cnt > 0.0f) ? 1.0f : 0.0f;
    __syncthreads();

    // tree-reduce the 64 labels of each batch
    for (int off = 32; off >= 1; off >>= 1) {
        if (l < off) {
            s_var [tid] += s_var [tid + off];
            s_pres[tid] += s_pres[tid + off];
        }
        __syncthreads();
    }

    if (tid == 0) {
        float loss = 0.0f;
        for (int bb = 0; bb < Bn; ++bb)
            loss += s_var[bb * Ln] / (s_pres[bb * Ln] + 1e-8f);
        out[0] = loss / (float)Bn;
    }
}

// ---------------------------------------------------------------------------
extern "C" void kernel_launch(void* const* d_in, const int* in_sizes, int n_in,
                              void* d_out, int out_size, void* d_ws, size_t ws_size,
                              hipStream_t stream) {
    const float* x = (const float*)d_in[0];
    const int*   t = (const int*)d_in[1];
    float* ws   = (float*)d_ws;
    float* gcnt = ws;
    float* gsum = ws + WS_CNT;
    float* gss  = ws + WS_CNT + WS_MOM;

    lv_zero<<<(WS_TOTAL + 255) / 256, 256, 0, stream>>>(ws, WS_TOTAL);
    lv_accum<<<Bn * WGS_PER_B, BLOCK, 0, stream>>>(x, t, gcnt, gsum, gss);
    lv_final<<<1, Bn * Ln, 0, stream>>>(gcnt, gsum, gss, (float*)d_out);
}